// GlobalClassGatherModule_38577396253240
// MI455X (gfx1250) — compile-verified
//
#include <hip/hip_runtime.h>
#include <hip/hip_bf16.h>
#include <math.h>
#include <stdint.h>

// Problem constants (reference: B,C,H,W,K = 8,128,256,256,19)
static constexpr int B_  = 8;
static constexpr int C_  = 128;
static constexpr int K_  = 19;
static constexpr int KP_ = 32;      // classes padded to 32 for 16x16 WMMA tiles
static constexpr int N_  = 65536;   // H*W

typedef float v2f __attribute__((ext_vector_type(2)));
typedef float v8f __attribute__((ext_vector_type(8)));
typedef int   v4i __attribute__((ext_vector_type(4)));

#if defined(__gfx1250__) && __has_builtin(__builtin_amdgcn_wmma_f32_16x16x4_f32)
#define HAVE_WMMA_F32X4 1
#endif

#if defined(__gfx1250__) && \
    __has_builtin(__builtin_amdgcn_global_load_async_to_lds_b128) && \
    __has_builtin(__builtin_amdgcn_s_wait_asynccnt)
#define HAVE_ASYNC_LDS 1
// Builtin signature (from hipcc diagnostic): (int4 AS1*, int4 AS3*, int, int).
typedef __attribute__((address_space(1))) v4i* as1_v4i;
typedef __attribute__((address_space(3))) v4i* as3_v4i;
// Generic->AS1 is value-preserving; generic LDS pointer's low 32 bits are the
// LDS byte offset.
__device__ __forceinline__ void async_g2l_b128(const void* g, void* l) {
    const unsigned loff = (unsigned)(uintptr_t)l;
    __builtin_amdgcn_global_load_async_to_lds_b128(
        (as1_v4i)(uintptr_t)g, (as3_v4i)(uintptr_t)loff, 0, 0);
}
#endif

// Monotonic unsigned encoding of float for atomicMax; enc==0 acts as -inf sentinel.
__device__ __forceinline__ unsigned encf(float x) {
    unsigned u = __float_as_uint(x);
    return (u & 0x80000000u) ? ~u : (u | 0x80000000u);
}
__device__ __forceinline__ float decf(unsigned e) {
    unsigned u = (e & 0x80000000u) ? (e & 0x7FFFFFFFu) : ~e;
    return __uint_as_float(u);
}

// ---------------------------------------------------------------------------
// One 16x16 output tile of D += A(16xK) * B(Kx16), f32 WMMA (K multiple of 4).
// A/B supplied via element functors la(row,k), lb(k,col) (tile-local indices).
// Lane mapping follows CDNA5 ISA 7.12.2:
//   A 16x4: lanes 0-15 hold K=0,1 (V0,V1); lanes 16-31 hold K=2,3.
//   C/D:    VGPR r -> row r (+8 for lanes 16-31), col = lane&15.
// Unroll capped at 4 to keep B-fragment live ranges short (round-1 asm showed
// scratch spills from a fully-unrolled K=128 loop).
// ---------------------------------------------------------------------------
template <typename FA, typename FB>
__device__ __forceinline__ void mm_tile(FA la, FB lb, int Ksz, int lane, float acc[8]) {
#ifdef HAVE_WMMA_F32X4
    v8f c;
    #pragma unroll
    for (int i = 0; i < 8; ++i) c[i] = acc[i];
    const int half = lane >> 4, r = lane & 15;
    #pragma unroll 4
    for (int k0 = 0; k0 < Ksz; k0 += 4) {
        const int ka = k0 + 2 * half;
        v2f a, bb;
        a.x  = la(r, ka);      a.y  = la(r, ka + 1);
        bb.x = lb(ka, r);      bb.y = lb(ka + 1, r);
        // 8 args: (neg_a, A, neg_b, B, c_mod, C, reuse_a, reuse_b)
        c = __builtin_amdgcn_wmma_f32_16x16x4_f32(false, a, false, bb, (short)0, c,
                                                  false, false);
    }
    #pragma unroll
    for (int i = 0; i < 8; ++i) acc[i] = c[i];
#else
    // Scalar fallback (also used for the host compilation pass).
    const int half = lane >> 4, cn = lane & 15;
    for (int r = 0; r < 8; ++r) {
        const int row = r + half * 8;
        float s = acc[r];
        for (int k = 0; k < Ksz; ++k) s += la(row, k) * lb(k, cn);
        acc[r] = s;
    }
#endif
}

// ---------------------------------------------------------------------------
// Kernel 1: per-pixel argmax class, its logit value, per-class max & count.
// ---------------------------------------------------------------------------
__global__ void k_argmax(const float* __restrict__ preds, unsigned char* __restrict__ cls,
                         float* __restrict__ pval, unsigned* __restrict__ mxenc,
                         float* __restrict__ cnt) {
    __shared__ unsigned smax[K_];
    __shared__ unsigned scnt[K_];
    const int t = threadIdx.x;
    if (t < K_) { smax[t] = 0u; scnt[t] = 0u; }
    __syncthreads();

    const long long gid = (long long)blockIdx.x * 256 + t;   // b*N + n
    const int b = (int)(gid >> 16);
    const int n = (int)(gid & (N_ - 1));
    const float* p = preds + (size_t)b * K_ * N_ + n;

    float best = p[0];
    int   bi   = 0;
    #pragma unroll
    for (int k = 1; k < K_; ++k) {
        const float v = p[(size_t)k * N_];
        if (v > best) { best = v; bi = k; }   // strict > keeps first (jnp.argmax)
    }
    cls[gid]  = (unsigned char)bi;
    pval[gid] = best;
    atomicMax(&smax[bi], encf(best));
    atomicAdd(&scnt[bi], 1u);
    __syncthreads();

    if (t < K_) {
        atomicMax(&mxenc[b * KP_ + t], smax[t]);
        atomicAdd(&cnt[b * KP_ + t], (float)scnt[t]);
    }
}

// ---------------------------------------------------------------------------
// Kernel 2: single pass over x (256 MB): per-class Z, weighted sums WX and
// plain sums G via LDS accumulators (2-way replicated by lane parity to halve
// same-address ds_add_f32 serialization), flushed with global f32 atomics.
// One block = 1024 pixels (4 per thread, float4 x loads), 64 blocks per batch.
// ---------------------------------------------------------------------------
__global__ void k_accum(const float* __restrict__ x, const unsigned char* __restrict__ cls,
                        const float* __restrict__ pval, const unsigned* __restrict__ mxenc,
                        float* __restrict__ WX, float* __restrict__ G,
                        float* __restrict__ Z) {
    constexpr int REP = 2;
    __shared__ float aWX[REP][K_ * C_];
    __shared__ float aG[REP][K_ * C_];
    __shared__ float zacc[K_];
    __shared__ float smx[K_];
    const int t   = threadIdx.x;
    const int b   = blockIdx.x >> 6;
    const int blk = blockIdx.x & 63;
    const int rp  = t & (REP - 1);

    for (int i = t; i < K_ * C_; i += 256) {
        aWX[0][i] = 0.f; aWX[1][i] = 0.f;
        aG[0][i]  = 0.f; aG[1][i]  = 0.f;
    }
    if (t < K_) {
        zacc[t] = 0.f;
        const unsigned e = mxenc[b * KP_ + t];
        smx[t] = (e == 0u) ? 0.f : decf(e);   // empty-class guard (matches ref)
    }
    __syncthreads();

    const int n0 = blk * 1024;
    int   kc[4];
    float ev[4];
    const size_t pbase = (size_t)b * N_ + n0 + 4 * t;
    #pragma unroll
    for (int j = 0; j < 4; ++j) {
        kc[j] = cls[pbase + j];
        ev[j] = __expf(pval[pbase + j] - smx[kc[j]]);
        atomicAdd(&zacc[kc[j]], ev[j]);
    }

    const float4* xp4 = (const float4*)(x + (size_t)b * C_ * N_ + n0);
    for (int c = 0; c < C_; ++c) {
#if defined(__gfx1250__)
        if (c + 4 < C_)  // global_prefetch_b8 a few lines ahead of the stream
            __builtin_prefetch(xp4 + (size_t)(c + 4) * (N_ / 4) + t, 0, 1);
#endif
        const float4 xv = xp4[(size_t)c * (N_ / 4) + t];
        atomicAdd(&aWX[rp][kc[0] * C_ + c], ev[0] * xv.x);
        atomicAdd(&aG [rp][kc[0] * C_ + c], xv.x);
        atomicAdd(&aWX[rp][kc[1] * C_ + c], ev[1] * xv.y);
        atomicAdd(&aG [rp][kc[1] * C_ + c], xv.y);
        atomicAdd(&aWX[rp][kc[2] * C_ + c], ev[2] * xv.z);
        atomicAdd(&aG [rp][kc[2] * C_ + c], xv.z);
        atomicAdd(&aWX[rp][kc[3] * C_ + c], ev[3] * xv.w);
        atomicAdd(&aG [rp][kc[3] * C_ + c], xv.w);
    }
    __syncthreads();

    for (int i = t; i < K_ * C_; i += 256) {
        const int k = i / C_, c = i - k * C_;
        atomicAdd(&WX[(size_t)b * KP_ * C_ + k * C_ + c], aWX[0][i] + aWX[1][i]);
        atomicAdd(&G [(size_t)b * C_ * KP_ + c * KP_ + k], aG[0][i] + aG[1][i]);
    }
    if (t < K_) atomicAdd(&Z[b * KP_ + t], zacc[t]);
}

// ---------------------------------------------------------------------------
// Kernel 3 (WMMA): per batch, three small f32 GEMMs:
//   cf2[32x128] = CF[32x128] * w2^T + b2        (CF = WX / Z)
//   S  [128x32] = w1[128x128] * G[128x32] + b1 (x) cnt
//   gc [128x128]= S * cf2 * C^-0.5  -> d_out
// WX/G staging uses GLOBAL_LOAD_ASYNC_TO_LDS_B128 (ASYNCcnt) when available;
// WX is normalized by Z in-place in LDS after s_wait_asynccnt.
// ---------------------------------------------------------------------------
__global__ void k_gemms(const float* __restrict__ WXg, const float* __restrict__ Gg,
                        const float* __restrict__ Zg, const float* __restrict__ Cg,
                        const float* __restrict__ w1, const float* __restrict__ b1,
                        const float* __restrict__ w2, const float* __restrict__ b2,
                        float* __restrict__ out) {
    __shared__ float sBufA[KP_ * C_];   // phase 1: CF;  phase 2+: S[128][32]
    __shared__ float sG[C_ * KP_];      // G[c][k]
    __shared__ float sCf2[KP_ * C_];    // cf2[k][d]
    __shared__ float sZ[KP_], sCnt[KP_];

    const int t = threadIdx.x, b = blockIdx.x;
    const int lane = t & 31, wave = t >> 5;

    if (t < KP_) { sZ[t] = Zg[b * KP_ + t]; sCnt[t] = Cg[b * KP_ + t]; }

#ifdef HAVE_ASYNC_LDS
    // Async DMA: 2 x 16KB global -> LDS, 4 x b128 per thread per buffer.
    {
        const float* gG = Gg  + (size_t)b * (C_ * KP_);
        const float* gW = WXg + (size_t)b * (KP_ * C_);
        #pragma unroll
        for (int it = 0; it < 4; ++it) {
            const int off = it * 1024 + t * 4;           // float index, 16B chunks
            async_g2l_b128(gG + off, &sG[off]);
            async_g2l_b128(gW + off, &sBufA[off]);
        }
        __builtin_amdgcn_s_wait_asynccnt(0);
    }
    __syncthreads();
    // Normalize WX by Z in place; zero the padded class rows.
    for (int i = t; i < KP_ * C_; i += 256) {
        const int k = i >> 7;                            // i = k*128 + c
        sBufA[i] = (k < K_) ? sBufA[i] / fmaxf(sZ[k], 1e-30f) : 0.f;
    }
#else
    __syncthreads();
    for (int i = t; i < KP_ * C_; i += 256) {
        sG[i] = Gg[(size_t)b * (C_ * KP_) + i];          // same [c][k] layout
        const int k = i >> 7;                            // i = k*128 + c
        sBufA[i] = (k < K_) ? WXg[(size_t)b * (KP_ * C_) + i] / fmaxf(sZ[k], 1e-30f)
                            : 0.f;
    }
#endif
    __syncthreads();

    // GEMM1: cf2 = CF(32x128) x w2^T(128x128) + b2 ; 2x8 tiles, 2 per wave.
    for (int tt = wave; tt < 16; tt += 8) {
        const int m0 = (tt >> 3) * 16, n0 = (tt & 7) * 16;
        float acc[8] = {};
        mm_tile([&](int r, int k) { return sBufA[(m0 + r) * C_ + k]; },
                [&](int k, int cn) { return w2[(size_t)(n0 + cn) * C_ + k]; },
                C_, lane, acc);
        const int half = lane >> 4, cn = lane & 15;
        #pragma unroll
        for (int r = 0; r < 8; ++r) {
            const int row = m0 + r + half * 8, col = n0 + cn;
            sCf2[row * C_ + col] = acc[r] + b2[col];
        }
    }
    __syncthreads();

    // GEMM2: S = w1(128x128) x G(128x32) + b1*cnt ; 8x2 tiles, 2 per wave.
    for (int tt = wave * 2; tt < wave * 2 + 2; ++tt) {
        const int m0 = (tt >> 1) * 16, n0 = (tt & 1) * 16;
        float acc[8] = {};
        mm_tile([&](int r, int k) { return w1[(size_t)(m0 + r) * C_ + k]; },
                [&](int k, int cn) { return sG[k * KP_ + n0 + cn]; },
                C_, lane, acc);
        const int half = lane >> 4, cn = lane & 15;
        #pragma unroll
        for (int r = 0; r < 8; ++r) {
            const int row = m0 + r + half * 8, col = n0 + cn;
            sBufA[row * KP_ + col] = acc[r] + b1[row] * sCnt[col];
        }
    }
    __syncthreads();

    // GEMM3: gc = S(128x32) x cf2(32x128) * C^-0.5 ; 8x8 tiles, 8 per wave.
    for (int i = 0; i < 8; ++i) {
        const int tt = wave * 8 + i;
        const int m0 = (tt >> 3) * 16, n0 = (tt & 7) * 16;
        float acc[8] = {};
        mm_tile([&](int r, int k) { return sBufA[(m0 + r) * KP_ + k]; },
                [&](int k, int cn) { return sCf2[k * C_ + n0 + cn]; },
                KP_, lane, acc);
        const int half = lane >> 4, cn = lane & 15;
        #pragma unroll
        for (int r = 0; r < 8; ++r) {
            const int row = m0 + r + half * 8, col = n0 + cn;
            out[(size_t)b * C_ * C_ + row * C_ + col] = acc[r] * 0.08838834764831845f;
        }
    }
}

// ---------------------------------------------------------------------------
// Kernel 4: in-place row softmax over the last dim (128), one block per row.
// ---------------------------------------------------------------------------
__global__ void k_softmax(float* __restrict__ out) {
    __shared__ float red[C_];
    const int t = threadIdx.x;
    const size_t row = (size_t)blockIdx.x * C_;
    const float v = out[row + t];
    red[t] = v;
    __syncthreads();
    for (int s = C_ / 2; s > 0; s >>= 1) {
        if (t < s) red[t] = fmaxf(red[t], red[t + s]);
        __syncthreads();
    }
    const float mx = red[0];
    __syncthreads();
    const float e = __expf(v - mx);
    red[t] = e;
    __syncthreads();
    for (int s = C_ / 2; s > 0; s >>= 1) {
        if (t < s) red[t] += red[t + s];
        __syncthreads();
    }
    out[row + t] = e / red[0];
}

// ---------------------------------------------------------------------------
extern "C" void kernel_launch(void* const* d_in, const int* in_sizes, int n_in,
                              void* d_out, int out_size, void* d_ws, size_t ws_size,
                              hipStream_t stream) {
    (void)in_sizes; (void)n_in; (void)out_size; (void)ws_size;
    const float* x     = (const float*)d_in[0];
    const float* preds = (const float*)d_in[1];
    const float* w1    = (const float*)d_in[2];
    const float* b1    = (const float*)d_in[3];
    const float* w2    = (const float*)d_in[4];
    const float* b2    = (const float*)d_in[5];
    float* out = (float*)d_out;

    // Workspace layout (floats unless noted):
    //   WX [B][KP][C] | G [B][C][KP] | Z [B][KP] | CNT [B][KP] | MXENC [B][KP] u32
    //   PVAL [B][N] | CLS [B][N] u8
    float*    WX    = (float*)d_ws;
    float*    G     = WX + (size_t)B_ * KP_ * C_;
    float*    Z     = G + (size_t)B_ * C_ * KP_;
    float*    CNT   = Z + (size_t)B_ * KP_;
    unsigned* MXENC = (unsigned*)(CNT + (size_t)B_ * KP_);
    float*    PVAL  = (float*)(MXENC + (size_t)B_ * KP_);
    unsigned char* CLS = (unsigned char*)(PVAL + (size_t)B_ * N_);

    const size_t zero_bytes =
        ((size_t)B_ * KP_ * C_ * 2 + (size_t)B_ * KP_ * 3) * sizeof(float);
    (void)hipMemsetAsync(d_ws, 0, zero_bytes, stream);

    k_argmax<<<dim3((B_ * N_) / 256), dim3(256), 0, stream>>>(preds, CLS, PVAL, MXENC, CNT);
    k_accum<<<dim3(B_ * 64), dim3(256), 0, stream>>>(x, CLS, PVAL, MXENC, WX, G, Z);
    k_gemms<<<dim3(B_), dim3(256), 0, stream>>>(WX, G, Z, CNT, w1, b1, w2, b2, out);
    k_softmax<<<dim3(B_ * C_), dim3(C_), 0, stream>>>(out);
}